// DFConv_59674275611059
// MI455X (gfx1250) — compile-verified
//
#include <hip/hip_runtime.h>
#include <math.h>

// Problem constants (match reference)
#define BB 8
#define CC 128
#define OO 128
#define HH 64
#define WW 64
#define HWSZ 4096          // H*W
#define KKT 9              // 3*3 taps
#define NOFF 18            // 2*KK offset channels
#define PRPITCH 144        // LDS pair-row pitch in float2 units (bank-conflict free)

typedef float v2f __attribute__((ext_vector_type(2)));
typedef float v8f __attribute__((ext_vector_type(8)));

// ---------------------------------------------------------------------------
// Kernel 1: transpose + pair-interleave w_dcn (O,C,3,3) ->
//   Wt2[k][c/2][o] = float2{ W[o][c][k], W[o][c+1][k] }   (147456 floats)
// so the main kernel's B-fragment is a single ds_load_b64.
// ---------------------------------------------------------------------------
__global__ __launch_bounds__(256)
void dfc_transpose_w(const float* __restrict__ wd, float* __restrict__ wt) {
    int i = blockIdx.x * 256 + threadIdx.x;      // i = o*1152 + c*9 + k
    if (i >= OO * CC * KKT) return;
    int o = i / (CC * KKT);
    int r = i - o * (CC * KKT);
    int c = r / KKT;
    int k = r - c * KKT;
    // layout: k * (64*128*2) + (c>>1)*(128*2) + o*2 + (c&1)
    wt[k * (CC / 2 * OO * 2) + (c >> 1) * (OO * 2) + o * 2 + (c & 1)] = wd[i];
}

// ---------------------------------------------------------------------------
// Kernel 2: offset conv.  One thread per pixel, 18 accumulators.
// w_off staged in LDS as [c][tap][oc(pad 20)] so weights read as b128/b64.
// ---------------------------------------------------------------------------
__global__ __launch_bounds__(256)
void dfc_offset_conv(const float* __restrict__ x, const float* __restrict__ wof,
                     const float* __restrict__ bof, float* __restrict__ offs) {
    __shared__ float lw[CC * KKT * 20];          // 92160 B, padding never read
    const int tid = threadIdx.x;
    for (int i = tid; i < NOFF * CC * KKT; i += 256) {
        int oc = i / (CC * KKT);                 // wof layout: oc*1152 + c*9 + t
        int r  = i - oc * (CC * KKT);
        lw[r * 20 + oc] = wof[i];
    }
    __syncthreads();

    const int m  = blockIdx.x * 256 + tid;       // pixel id
    const int b  = m >> 12;
    const int hw = m & 4095;
    const int h  = hw >> 6;
    const int w  = hw & 63;

    float acc[NOFF];
#pragma unroll
    for (int oc = 0; oc < NOFF; ++oc) acc[oc] = bof[oc];

    const float* xb = x + (size_t)b * CC * HWSZ;
    for (int c = 0; c < CC; ++c) {
        const float* xc = xb + c * HWSZ;
#pragma unroll
        for (int t = 0; t < KKT; ++t) {
            const int yy = h + t / 3 - 1;
            const int xx = w + t % 3 - 1;
            float xv = 0.f;
            if (yy >= 0 && yy < HH && xx >= 0 && xx < WW)
                xv = xc[yy * WW + xx];
            const float* lt = lw + (c * KKT + t) * 20;
            const float4 w0 = *(const float4*)(lt);
            const float4 w1 = *(const float4*)(lt + 4);
            const float4 w2 = *(const float4*)(lt + 8);
            const float4 w3 = *(const float4*)(lt + 12);
            const float2 w4 = *(const float2*)(lt + 16);
            acc[0]  += xv * w0.x;  acc[1]  += xv * w0.y;
            acc[2]  += xv * w0.z;  acc[3]  += xv * w0.w;
            acc[4]  += xv * w1.x;  acc[5]  += xv * w1.y;
            acc[6]  += xv * w1.z;  acc[7]  += xv * w1.w;
            acc[8]  += xv * w2.x;  acc[9]  += xv * w2.y;
            acc[10] += xv * w2.z;  acc[11] += xv * w2.w;
            acc[12] += xv * w3.x;  acc[13] += xv * w3.y;
            acc[14] += xv * w3.z;  acc[15] += xv * w3.w;
            acc[16] += xv * w4.x;  acc[17] += xv * w4.y;
        }
    }
    float* ob = offs + (size_t)b * NOFF * HWSZ + hw;
#pragma unroll
    for (int oc = 0; oc < NOFF; ++oc) ob[oc * HWSZ] = acc[oc];
}

// ---------------------------------------------------------------------------
// Kernel 3: fused bilinear-sample + GEMM via V_WMMA_F32_16X16X4_F32.
// Wave = 16-pixel M-tile x full N=128 (8 accumulators).
// K flattened c-fastest (kk = k*128 + c): each x4 chunk = one tap, 4 channels,
// so sampling geometry is computed once per tap and reused for 32 chunks.
// B-fragment = one ds_load_b64 per N-tile (pair-interleaved weights).
// ---------------------------------------------------------------------------
__global__ __launch_bounds__(256)
void dfc_main_wmma(const float* __restrict__ x, const float* __restrict__ offs,
                   const float* __restrict__ wt, float* __restrict__ out) {
    __shared__ float2 lw2[(CC / 2) * PRPITCH];   // 73728 B: Wt2[k] slice, padded
    const int tid  = threadIdx.x;
    const int lane = tid & 31;
    const int wid  = tid >> 5;
    const int half = lane >> 4;                  // 0: K-pair {0,1}, 1: {2,3}
    const int l16  = lane & 15;

    const int tile = blockIdx.x * 8 + wid;       // 2048 M-tiles total
    const int m  = tile * 16 + l16;              // this lane's A-row pixel
    const int b  = m >> 12;
    const int hw = m & 4095;
    const int h  = hw >> 6;
    const int w  = hw & 63;

    const float* xb_base  = x    + (size_t)b * CC * HWSZ;
    const float* off_base = offs + (size_t)b * NOFF * HWSZ + hw;

    v8f acc[8];
#pragma unroll
    for (int nt = 0; nt < 8; ++nt)
#pragma unroll
        for (int r = 0; r < 8; ++r) acc[nt][r] = 0.f;

    for (int k = 0; k < KKT; ++k) {
        // ---- stage Wt2[k] (64 pair-rows x 128 float2) into LDS, pitch 144 ----
        __syncthreads();
        const float2* src2 = (const float2*)(wt + k * CC * OO);
        for (int i = tid; i < (CC / 2) * OO; i += 256)
            lw2[(i >> 7) * PRPITCH + (i & 127)] = src2[i];
        __syncthreads();

        // ---- per-tap bilinear geometry (branchless, per lane) ----
        const int   ky = k / 3 - 1, kx = k % 3 - 1;
        const float dy = off_base[(2 * k)     * HWSZ];
        const float dx = off_base[(2 * k + 1) * HWSZ];
        const float ys = (float)(h + ky) + dy;
        const float xs = (float)(w + kx) + dx;
        const float y0f = floorf(ys), x0f = floorf(xs);
        const float wy = ys - y0f,  wx = xs - x0f;
        const int y0 = (int)y0f, x0 = (int)x0f;
        const int y1 = y0 + 1,   x1 = x0 + 1;
        const float vy0 = (y0 >= 0 && y0 < HH) ? 1.f : 0.f;
        const float vy1 = (y1 >= 0 && y1 < HH) ? 1.f : 0.f;
        const float vx0 = (x0 >= 0 && x0 < WW) ? 1.f : 0.f;
        const float vx1 = (x1 >= 0 && x1 < WW) ? 1.f : 0.f;
        const int y0c = min(max(y0, 0), HH - 1), y1c = min(max(y1, 0), HH - 1);
        const int x0c = min(max(x0, 0), WW - 1), x1c = min(max(x1, 0), WW - 1);
        const int o00 = y0c * WW + x0c, o01 = y0c * WW + x1c;
        const int o10 = y1c * WW + x0c, o11 = y1c * WW + x1c;
        const float f00 = (1.f - wy) * (1.f - wx) * vy0 * vx0;
        const float f01 = (1.f - wy) * wx         * vy0 * vx1;
        const float f10 = wy * (1.f - wx)         * vy1 * vx0;
        const float f11 = wy * wx                 * vy1 * vx1;

        // ---- 32 K-chunks of 4 channels; 8 WMMAs + 8 ds_load_b64 each ----
        for (int cq = 0; cq < CC / 4; ++cq) {
            const int cpair = cq * 4 + half * 2;
            const float* p0 = xb_base + cpair * HWSZ;
            const float* p1 = p0 + HWSZ;
            v2f a;
            a.x = f00 * p0[o00] + f01 * p0[o01] + f10 * p0[o10] + f11 * p0[o11];
            a.y = f00 * p1[o00] + f01 * p1[o01] + f10 * p1[o10] + f11 * p1[o11];

            // pair-row index = cpair/2 = cq*2 + half
            const float2* lbp = lw2 + (cq * 2 + half) * PRPITCH + l16;
#pragma unroll
            for (int nt = 0; nt < 8; ++nt) {
                const float2 wv = lbp[nt * 16];  // {W[cpair][n], W[cpair+1][n]}
                v2f bf;
                bf.x = wv.x;
                bf.y = wv.y;
                acc[nt] = __builtin_amdgcn_wmma_f32_16x16x4_f32(
                    false, a, false, bf, (short)0, acc[nt], false, false);
            }
        }
    }

    // ---- store: lane holds n = nt*16 + l16; VGPR r -> pixel tile*16 + 8*half + r
    const int m0 = tile * 16 + 8 * half;
#pragma unroll
    for (int nt = 0; nt < 8; ++nt) {
        const int n = nt * 16 + l16;
#pragma unroll
        for (int r = 0; r < 8; ++r) {
            const int mm = m0 + r;
            const int bb = mm >> 12;
            const int phw = mm & 4095;
            out[(((size_t)bb * OO + n) << 12) + phw] = acc[nt][r];
        }
    }
}

// ---------------------------------------------------------------------------
extern "C" void kernel_launch(void* const* d_in, const int* in_sizes, int n_in,
                              void* d_out, int out_size, void* d_ws, size_t ws_size,
                              hipStream_t stream) {
    (void)in_sizes; (void)n_in; (void)out_size; (void)ws_size;
    const float* x   = (const float*)d_in[0];
    const float* wof = (const float*)d_in[1];
    const float* bof = (const float*)d_in[2];
    const float* wd  = (const float*)d_in[3];
    float* out  = (float*)d_out;

    float* offs = (float*)d_ws;                          // 589,824 f32 (2.4 MB)
    float* wtr  = offs + (size_t)BB * NOFF * HWSZ;       // 147,456  f32 (0.6 MB)

    dfc_transpose_w<<<(OO * CC * KKT + 255) / 256, 256, 0, stream>>>(wd, wtr);
    dfc_offset_conv<<<(BB * HWSZ) / 256, 256, 0, stream>>>(x, wof, bof, offs);
    dfc_main_wmma<<<(BB * HWSZ) / (16 * 8), 256, 0, stream>>>(x, offs, wtr, out);
}